// HarmonicOscillator_5669356835580
// MI455X (gfx1250) — compile-verified
//
#include <hip/hip_runtime.h>
#include <hip/hip_bf16.h>
#include <stdint.h>

#define SR_F     16000.0f
#define T_FRAMES 1000
#define P_PART   64
#define FRAME    64
#define NYQ      8000.0f
#define GAIN     0.02f

// ---------------------------------------------------------------------------
// Kernel 1: per-batch exclusive scan of per-frame phase increments.
// Frame sum of upsampled f0 is analytic: S[t] = 8*f0[t-1] + 48*f0[t] + 8*f0[t+1]
// (indices clamped). phi0[b,t] = (sum_{t'<t} S[b,t']) / SR.
// ---------------------------------------------------------------------------
__global__ void __launch_bounds__(256)
frame_phase_scan(const float* __restrict__ f0, float* __restrict__ phi0) {
  const int b   = blockIdx.x;
  const int tid = threadIdx.x;
  const float* f = f0 + (size_t)b * T_FRAMES;

  float loc[4];
  float tot = 0.f;
#pragma unroll
  for (int k = 0; k < 4; ++k) {
    int t = tid * 4 + k;
    float v = 0.f;
    if (t < T_FRAMES) {
      int tm = (t > 0) ? t - 1 : 0;
      int tp = (t < T_FRAMES - 1) ? t + 1 : T_FRAMES - 1;
      v = 8.f * f[tm] + 48.f * f[t] + 8.f * f[tp];
    }
    loc[k] = tot;   // exclusive prefix within this thread
    tot += v;
  }

  __shared__ float sc[256];
  sc[tid] = tot;
  __syncthreads();
  // Hillis-Steele inclusive scan over thread totals
  for (int off = 1; off < 256; off <<= 1) {
    float v = (tid >= off) ? sc[tid - off] : 0.f;
    __syncthreads();
    sc[tid] += v;
    __syncthreads();
  }
  float excl = (tid == 0) ? 0.f : sc[tid - 1];

#pragma unroll
  for (int k = 0; k < 4; ++k) {
    int t = tid * 4 + k;
    if (t < T_FRAMES)
      phi0[(size_t)b * T_FRAMES + t] = (excl + loc[k]) * (1.f / SR_F);
  }
}

// ---------------------------------------------------------------------------
// Kernel 2: one block per (batch, group of 4 frames); one thread per output
// sample. Stages 6 c-columns into LDS via gfx1250 async global->LDS, applies
// the anti-alias mask in LDS once, then runs the 64-partial sine loop using
// hardware v_sin_f32 (argument already in revolutions).
// ---------------------------------------------------------------------------
__global__ void __launch_bounds__(256)
synth_kernel(const float* __restrict__ f0, const float* __restrict__ c,
             const float* __restrict__ a, const float* __restrict__ phi0,
             float* __restrict__ out) {
  const int b   = blockIdx.y;
  const int t0  = blockIdx.x * 4;      // first frame handled by this block
  const int tid = threadIdx.x;         // 0..255
  const int fbk = tid >> 6;            // frame within block: 0..3
  const int j   = tid & 63;            // sample within frame: 0..63
  const int t   = t0 + fbk;

  __shared__ float lds_c[6 * P_PART];  // columns (t0-1 .. t0+4), clamped

  const size_t cb  = (size_t)b * P_PART * T_FRAMES;
  const float* fb0 = f0 + (size_t)b * T_FRAMES;

  // ---- stage 384 elements: e = tid for all, e = tid+256 for tid < 128 ----
  {
    int e = tid;
    int s = e >> 6, p = e & 63;
    int col = t0 - 1 + s;
    col = col < 0 ? 0 : (col > T_FRAMES - 1 ? T_FRAMES - 1 : col);
    const float* g = c + cb + (size_t)p * T_FRAMES + (size_t)col;
    uint32_t loff = (uint32_t)(uintptr_t)(&lds_c[e]);  // flat low 32 bits = LDS offset
    asm volatile("global_load_async_to_lds_b32 %0, %1, off"
                 :: "v"(loff), "v"(g) : "memory");
  }
  if (tid < 128) {
    int e = tid + 256;
    int s = e >> 6, p = e & 63;
    int col = t0 - 1 + s;
    col = col < 0 ? 0 : (col > T_FRAMES - 1 ? T_FRAMES - 1 : col);
    const float* g = c + cb + (size_t)p * T_FRAMES + (size_t)col;
    uint32_t loff = (uint32_t)(uintptr_t)(&lds_c[e]);
    asm volatile("global_load_async_to_lds_b32 %0, %1, off"
                 :: "v"(loff), "v"(g) : "memory");
  }
  asm volatile("s_wait_asynccnt 0" ::: "memory");
  __syncthreads();

  // ---- apply anti-alias mask in LDS (depends only on (p, column)) ----
  {
    int e = tid;
    int s = e >> 6, p = e & 63;
    int col = t0 - 1 + s;
    col = col < 0 ? 0 : (col > T_FRAMES - 1 ? T_FRAMES - 1 : col);
    if (fb0[col] * (float)(p + 1) >= NYQ) lds_c[e] = 0.f;
  }
  if (tid < 128) {
    int e = tid + 256;
    int s = e >> 6, p = e & 63;
    int col = t0 - 1 + s;
    col = col < 0 ? 0 : (col > T_FRAMES - 1 ? T_FRAMES - 1 : col);
    if (fb0[col] * (float)(p + 1) >= NYQ) lds_c[e] = 0.f;
  }
  __syncthreads();

  // ---- per-sample closed-form phase + interpolation setup ----
  const float fprev = fb0[(t > 0) ? t - 1 : 0];
  const float fcur  = fb0[t];
  const float fnext = fb0[(t < T_FRAMES - 1) ? t + 1 : T_FRAMES - 1];
  const float* ab   = a + (size_t)b * T_FRAMES;

  float w, P, la, lb;
  int slotA, slotB;
  if (j < 32) {
    // src = t - (31.5 - j)/64, interpolates (t-1, t); w = (j + 32.5)/64
    w = ((float)j + 32.5f) * (1.f / 64.f);
    float jp1 = (float)(j + 1);
    float Sw  = jp1 * ((float)j + 65.f) * (1.f / 128.f);  // sum of weights 0..j
    P = fprev * (jp1 - Sw) + fcur * Sw;                   // inclusive in-frame prefix
    slotA = fbk;  slotB = fbk + 1;
    la = ab[(t > 0) ? t - 1 : 0];  lb = ab[t];
  } else {
    // interpolates (t, t+1); w = (j - 31.5)/64
    w = ((float)j - 31.5f) * (1.f / 64.f);
    float Kp1 = (float)(j - 31);
    P = 8.f * fprev + 24.f * fcur + Kp1 * fcur +
        (fnext - fcur) * (Kp1 * Kp1) * (1.f / 128.f);
    slotA = fbk + 1;  slotB = fbk + 2;
    la = ab[t];  lb = ab[(t < T_FRAMES - 1) ? t + 1 : T_FRAMES - 1];
  }
  const float wB = w, wA = 1.f - w;
  const float loud = la * wA + lb * wB;

  float Phi = phi0[(size_t)b * T_FRAMES + t] + P * (1.f / SR_F);
  float phi = Phi - floorf(Phi);   // reduce once; per-partial arg = (p+1)*phi <= 64

  const float* cA = &lds_c[slotA * P_PART];
  const float* cB = &lds_c[slotB * P_PART];
  float acc = 0.f;
  float pm  = 0.f;
#pragma unroll 16
  for (int p = 0; p < P_PART; ++p) {
    pm += 1.f;
    float aaud = fmaf(cA[p], wA, cB[p] * wB);         // broadcast LDS reads
    float s    = __builtin_amdgcn_sinf(pm * phi);     // v_sin_f32: sin(2*pi*x)
    acc = fmaf(s, aaud, acc);
  }

  out[(size_t)b * (T_FRAMES * FRAME) + (size_t)t * FRAME + j] = acc * GAIN * loud;
}

// ---------------------------------------------------------------------------
extern "C" void kernel_launch(void* const* d_in, const int* in_sizes, int n_in,
                              void* d_out, int out_size, void* d_ws, size_t ws_size,
                              hipStream_t stream) {
  (void)n_in; (void)out_size; (void)ws_size;
  const float* f0 = (const float*)d_in[0];   // (B, 1000)
  const float* c  = (const float*)d_in[1];   // (B, 64, 1000)
  const float* a  = (const float*)d_in[2];   // (B, 1000)
  float* out      = (float*)d_out;           // (B, 64000)
  const int B     = in_sizes[0] / T_FRAMES;

  float* phi0 = (float*)d_ws;                // B*1000 floats of scratch

  frame_phase_scan<<<B, 256, 0, stream>>>(f0, phi0);

  dim3 grid(T_FRAMES / 4, B);
  synth_kernel<<<grid, 256, 0, stream>>>(f0, c, a, phi0, out);
}